// CrossHeadProjectionV2_32384053412068
// MI455X (gfx1250) — compile-verified
//
#include <hip/hip_runtime.h>

// CDNA5 / gfx1250, wave32.
typedef __attribute__((ext_vector_type(2))) float v2f;
typedef __attribute__((ext_vector_type(8))) float v8f;

#define TT 2048
#define SS 2048
#define NHEAD 16
#define NS_ITER 4   // s-tiles (of 16) per wave

// ---------------------------------------------------------------------------
// Prep: A[t][m][n] = w[m][n] + (m==n)*(1 + qdd[t][m])
//                  + sum_i qw1[t][i][m]*qw2[t][i][n]
// so that ret[n] = sum_m A[m][n] * x[m]  covers static mixing, query-side
// low-rank and the qdd diagonal. 2048 x 16 x 16 floats = 2 MB in workspace.
// ---------------------------------------------------------------------------
__global__ __launch_bounds__(256) void chp_prep_A(
    const float* __restrict__ qw1, const float* __restrict__ qw2,
    const float* __restrict__ qdd, const float* __restrict__ w,
    float* __restrict__ A) {
  int t  = blockIdx.x;
  int mn = threadIdx.x;       // 0..255
  int m  = mn >> 4;
  int n  = mn & 15;
  const float* q1 = qw1 + t * 32;   // [i][m], I=2, M=16
  const float* q2 = qw2 + t * 32;
  float a = w[mn];
  if (m == n) a += 1.0f + qdd[t * 16 + m];
  a = fmaf(q1[m],      q2[n],      a);
  a = fmaf(q1[16 + m], q2[16 + n], a);
  A[t * 256 + mn] = a;
}

// ---------------------------------------------------------------------------
// Main: per (t, 16-column s-tile) do D[n][s] = sum_m A_t[m][n] * x[m][s]
// via 4 chained V_WMMA_F32_16X16X4_F32, then add key-side rank-2 + kdd diag
// on the VALU (co-issues against the matrix pipe).
//
// WMMA f32 16x16x4 operand layout (ISA 7.12.2):
//   A 16x4:  lane L (<16): a.x=A[M=L][K=0], a.y=A[M=L][K=1]
//            lane L (>=16): a.x=A[M=L-16][K=2], a.y=A[M=L-16][K=3]
//   B 4x16:  (mirrored) lane L: b.x=B[K=2*hi][N=L%16], b.y=B[K=2*hi+1][N=L%16]
//   C/D:     VGPR v, lanes 0-15 -> (M=v, N=lane); lanes 16-31 -> (M=v+8)
// We compute D[M=n][N=s] = sum_K Awmma[n][K=m]*Bwmma[m][s],
// with Awmma[n][m] = A_t[m][n] and Bwmma[m][s] = x[m][t][s].
//
// x (read once) and out (written once) are streamed non-temporally so the
// 512 MB stream does not evict the hot A_t / kw1 / kw2 / kdd tables
// (~2.7 MB, reused 2048x) from the 192 MB L2.
// ---------------------------------------------------------------------------
__global__ __launch_bounds__(256) void chp_main(
    const float* __restrict__ x,   const float* __restrict__ A,
    const float* __restrict__ kw1, const float* __restrict__ kw2,
    const float* __restrict__ kdd, float* __restrict__ out) {
  const int lane = threadIdx.x & 31;
  const int wave = threadIdx.x >> 5;   // 0..7
  const int t    = blockIdx.y;
  const int col  = lane & 15;
  const int hi   = lane >> 4;          // which half-wave
  const int sbase = (blockIdx.x * 8 + wave) * (16 * NS_ITER);

  const size_t TS = (size_t)TT * SS;
  const float* At = A + t * 256;

  // Load the A_t fragments once; reused for all NS_ITER s-tiles.
  v2f av[4];
#pragma unroll
  for (int kk = 0; kk < 4; ++kk) {
    int m0 = 4 * kk + 2 * hi;
    av[kk].x = At[m0 * 16 + col];
    av[kk].y = At[(m0 + 1) * 16 + col];
  }

  const float* xt = x   + (size_t)t * SS;  // + m*TS + s
  float*       ot = out + (size_t)t * SS;

#pragma unroll
  for (int it = 0; it < NS_ITER; ++it) {
    const int s = sbase + it * 16 + col;

    v8f c = {0.f, 0.f, 0.f, 0.f, 0.f, 0.f, 0.f, 0.f};
    float xm[8];   // this lane's x[m][s] for m = 4*kk + 2*hi + {0,1}
#pragma unroll
    for (int kk = 0; kk < 4; ++kk) {
      const int m0 = 4 * kk + 2 * hi;
      v2f bv;
      bv.x = __builtin_nontemporal_load(&xt[(size_t)m0 * TS + s]);
      bv.y = __builtin_nontemporal_load(&xt[(size_t)(m0 + 1) * TS + s]);
      xm[2 * kk]     = bv.x;
      xm[2 * kk + 1] = bv.y;
      c = __builtin_amdgcn_wmma_f32_16x16x4_f32(false, av[kk], false, bv,
                                                (short)0, c, false, false);
    }

    // Reassemble the full 16-element x column in every lane: partner lane
    // (lane ^ 16) holds the other half of the K values for the same s column.
    float xcol[16];
#pragma unroll
    for (int kk = 0; kk < 4; ++kk) {
#pragma unroll
      for (int j = 0; j < 2; ++j) {
        float mine  = xm[2 * kk + j];
        float other = __shfl_xor(mine, 16, 32);
        xcol[4 * kk + j]     = hi ? other : mine;   // K = 4kk + {0,1}
        xcol[4 * kk + 2 + j] = hi ? mine  : other;  // K = 4kk + {2,3}
      }
    }

    // Key-side low-rank: hk[i] = sum_m kw1[s][i][m] * x[m][s]
    const float* k1 = kw1 + s * 32;   // [i][m]
    float hk0 = 0.f, hk1 = 0.f;
#pragma unroll
    for (int m = 0; m < 16; ++m) {
      hk0 = fmaf(k1[m],      xcol[m], hk0);
      hk1 = fmaf(k1[16 + m], xcol[m], hk1);
    }

    // This lane owns output rows n = 8*hi + v (v = 0..7), column s.
    const float* k2 = kw2 + s * 32 + 8 * hi;
    const float* kd = kdd + s * 16 + 8 * hi;
#pragma unroll
    for (int v = 0; v < 8; ++v) {
      float xn = hi ? xcol[8 + v] : xcol[v];
      float r  = c[v];
      r = fmaf(hk0, k2[v],      r);
      r = fmaf(hk1, k2[16 + v], r);
      r = fmaf(xn,  kd[v],      r);
      __builtin_nontemporal_store(r, &ot[(size_t)(8 * hi + v) * TS + s]);
    }
  }
}

// ---------------------------------------------------------------------------
// Launch
// ---------------------------------------------------------------------------
extern "C" void kernel_launch(void* const* d_in, const int* in_sizes, int n_in,
                              void* d_out, int out_size, void* d_ws, size_t ws_size,
                              hipStream_t stream) {
  const float* x   = (const float*)d_in[0];  // (1,16,2048,2048)
  const float* qw1 = (const float*)d_in[1];  // (1,2048,1,2,16)
  const float* qw2 = (const float*)d_in[2];
  const float* kw1 = (const float*)d_in[3];  // (1,2048,1,2,16)
  const float* kw2 = (const float*)d_in[4];
  const float* qdd = (const float*)d_in[5];  // (1,2048,1,16)
  const float* kdd = (const float*)d_in[6];
  const float* w   = (const float*)d_in[7];  // (1,16,16)
  float* out = (float*)d_out;
  float* A   = (float*)d_ws;                 // 2048*16*16 floats = 2 MB

  (void)in_sizes; (void)n_in; (void)out_size; (void)ws_size;

  chp_prep_A<<<dim3(TT), dim3(256), 0, stream>>>(qw1, qw2, qdd, w, A);

  // grid: x = S / (8 waves * 4 tiles * 16) = 4, y = T
  dim3 grid(SS / (8 * NS_ITER * 16), TT);
  chp_main<<<grid, dim3(256), 0, stream>>>(x, A, kw1, kw2, kdd, out);
}